// SimpleGrid_60687887892817
// MI455X (gfx1250) — compile-verified
//
#include <hip/hip_runtime.h>

// Trilinear interpolation of a 256x256x256x2 fp32 grid at N=2M points.
// Memory/gather-bound kernel tuned for MI455X (gfx1250, wave32):
//  - x coordinates staged into LDS via CDNA5 async global->LDS engine
//    (global_load_async_to_lds_b32 in saddr+voffset32 form, s_wait_asynccnt).
//  - 8 corner gathers per point as global_load_b64 (float2 = both channels)
//    off one base address with immediate byte offsets -> 8 outstanding
//    loads/lane; grid (134 MB) is resident in the 192 MB L2.
//  - Branchless FMA trilerp, validity mask; sigma -> out[0..N), alpha -> out[N..2N).

__global__ __launch_bounds__(256) void trilerp_kernel(
    const float* __restrict__ x, const float* __restrict__ grid,
    float* __restrict__ out, int n) {
  __shared__ float xs[3 * 256];

  const unsigned t = threadIdx.x;
  const unsigned total3m1 = 3u * (unsigned)n - 1u;       // last valid x element
  const unsigned blockBase = blockIdx.x * (3u * 256u);   // element index of slab

  // Async copy this block's 768 coordinate floats into LDS (coalesced):
  // lane t copies elements {t, t+256, t+512} of the block's x-slab.
  // GVS addressing: SGPR base = x, per-lane 32-bit byte offset.
#pragma unroll
  for (unsigned k = 0; k < 3; ++k) {
    unsigned gi = blockBase + k * 256u + t;
    gi = (gi > total3m1) ? total3m1 : gi;  // cheap 32-bit clamp (tail safety)
    const unsigned goff = gi * 4u;
    // Generic->LDS address: hardware uses addr[31:0] as the LDS byte offset.
    const unsigned lds_off = (unsigned)(unsigned long long)(&xs[k * 256u + t]);
    asm volatile("global_load_async_to_lds_b32 %0, %1, %2"
                 :: "v"(lds_off), "v"(goff), "s"(x)
                 : "memory");
  }
  asm volatile("s_wait_asynccnt 0" ::: "memory");
  __syncthreads();

  const int i = (int)(blockIdx.x * 256u + t);
  if (i >= n) return;

  const float px = xs[3 * t + 0];
  const float py = xs[3 * t + 1];
  const float pz = xs[3 * t + 2];

  // index = (x - lower) * res, lower = -1, res = 127.5
  const float ix = (px + 1.0f) * 127.5f;
  const float iy = (py + 1.0f) * 127.5f;
  const float iz = (pz + 1.0f) * 127.5f;

  const bool valid = (ix >= 0.0f) && (ix <= 255.0f) &&
                     (iy >= 0.0f) && (iy <= 255.0f) &&
                     (iz >= 0.0f) && (iz <= 255.0f);

  // i0 = clip(floor(index), 0, size-2); f = index - i0
  const float fx0 = fminf(fmaxf(floorf(ix), 0.0f), 254.0f);
  const float fy0 = fminf(fmaxf(floorf(iy), 0.0f), 254.0f);
  const float fz0 = fminf(fmaxf(floorf(iz), 0.0f), 254.0f);
  const int ix0 = (int)fx0;
  const int iy0 = (int)fy0;
  const int iz0 = (int)fz0;
  const float fx = ix - fx0;
  const float fy = iy - fy0;
  const float fz = iz - fz0;

  // Base corner index in float2 units: strides x:65536, y:256, z:1.
  const int b = (ix0 << 16) + (iy0 << 8) + iz0;
  const float2* __restrict__ G = reinterpret_cast<const float2*>(grid);

  // 8 gathers, all immediate offsets from one address -> 8 global_load_b64.
  const float2 c000 = G[b];
  const float2 c001 = G[b + 1];
  const float2 c010 = G[b + 256];
  const float2 c011 = G[b + 257];
  const float2 c100 = G[b + 65536];
  const float2 c101 = G[b + 65536 + 1];
  const float2 c110 = G[b + 65536 + 256];
  const float2 c111 = G[b + 65536 + 257];

  const float gx = 1.0f - fx, gy = 1.0f - fy, gz = 1.0f - fz;
  const float w00 = gy * gz;
  const float w01 = gy * fz;
  const float w10 = fy * gz;
  const float w11 = fy * fz;

  // Plane x = ix0
  const float s0 = fmaf(w00, c000.x, fmaf(w01, c001.x, fmaf(w10, c010.x, w11 * c011.x)));
  const float a0 = fmaf(w00, c000.y, fmaf(w01, c001.y, fmaf(w10, c010.y, w11 * c011.y)));
  // Plane x = ix0 + 1
  const float s1 = fmaf(w00, c100.x, fmaf(w01, c101.x, fmaf(w10, c110.x, w11 * c111.x)));
  const float a1 = fmaf(w00, c100.y, fmaf(w01, c101.y, fmaf(w10, c110.y, w11 * c111.y)));

  const float m = valid ? 1.0f : 0.0f;
  const float sigma = fmaf(gx, s0, fx * s1) * m;
  const float alpha = fmaf(gx, a0, fx * a1) * m;

  out[i] = sigma;
  out[n + i] = alpha;
}

extern "C" void kernel_launch(void* const* d_in, const int* in_sizes, int n_in,
                              void* d_out, int out_size, void* d_ws, size_t ws_size,
                              hipStream_t stream) {
  const float* x    = (const float*)d_in[0];   // [N,3] fp32
  const float* grid = (const float*)d_in[1];   // [256,256,256,2] fp32
  float* out = (float*)d_out;                  // [2N] fp32: sigma | alpha

  const int n = in_sizes[0] / 3;
  const int blocks = (n + 255) / 256;
  hipLaunchKernelGGL(trilerp_kernel, dim3(blocks), dim3(256), 0, stream,
                     x, grid, out, n);
}